// OptimalTransportLayer_68375879352963
// MI455X (gfx1250) — compile-verified
//
#include <hip/hip_runtime.h>

// Sinkhorn (optimal transport) for B independent 256x256 matrices — MI455X.
//  - One workgroup per matrix; K = exp(-gamma*(M - rowmin)) resident in LDS
//    (256x258 f32 = 258KB < 320KB/WGP). HBM traffic = read M + write P only
//    (~268MB -> ~11.5us roofline at 23.3 TB/s).
//  - Iterations in scaling-vector form: u = r/(K v); beta_j = v_j*(K^T u)_j;
//    early-stop on max|beta - c| <= eps; else v = c/(K^T u). Two 256-matvecs
//    per iteration instead of rewriting the 65K-element matrix.
//  - Matvecs via V_WMMA_F32_16X16X4_F32 (f32 needed for the 1e-6 tolerance),
//    vector broadcast into the B (resp. A) operand; TWO independent
//    accumulators per strip to break the WMMA->WMMA RAW chain.
//  - M loaded with GLOBAL_LOAD_ASYNC_TO_LDS_B64 (ASYNCcnt path), no VGPR
//    round-trip.

typedef __attribute__((ext_vector_type(2))) float v2f;
typedef __attribute__((ext_vector_type(8))) float v8f;

#define H_ 256
#define W_ 256
#define PITCH 258          // even: 8B-aligned b64 tile loads; 258%64==2 -> conflict-free A loads
#define NTHREADS 256       // 8 wave32
#define GAMMA_C 1.0f
#define EPS_C 1e-6f
#define MAXITERS_C 1000

__global__ __launch_bounds__(NTHREADS)
void sinkhorn_wmma_kernel(const float* __restrict__ M, float* __restrict__ P)
{
    extern __shared__ float smem[];
    float* K = smem;                       // H_*PITCH floats
    float* u = K + H_ * PITCH;             // H_ floats (also holds row sums)
    float* v = u + H_;                     // W_ floats
    float* t = v + W_;                     // W_ floats (col sums)
    unsigned* maxbits = (unsigned*)(t + W_);

    const int tid  = threadIdx.x;
    const int lane = tid & 31;
    const int wave = tid >> 5;             // 0..7
    const int hi   = lane >> 4;            // 0: lanes 0-15, 1: lanes 16-31
    const int lo   = lane & 15;
    const float rmarg = 1.0f / (float)H_;
    const float cmarg = 1.0f / (float)W_;

    const float* Mb = M + (size_t)blockIdx.x * (H_ * W_);
    float*       Pb = P + (size_t)blockIdx.x * (H_ * W_);

    // ---- async copy M -> LDS (GLOBAL_LOAD_ASYNC_TO_LDS_B64, GVS mode) ----
    {
        // LDS byte offset of K within this block's allocation.
        unsigned lbase =
            (unsigned)(uintptr_t)(__attribute__((address_space(3))) float*)K;
        #pragma unroll 4
        for (int it = 0; it < (H_ * W_) / (NTHREADS * 2); ++it) {
            int fidx = (it * NTHREADS + tid) * 2;        // 2 floats / thread / iter
            int row = fidx >> 8, col = fidx & 255;       // col is even
            unsigned loff = lbase + (unsigned)(row * PITCH + col) * 4u;
            unsigned goff = (unsigned)fidx * 4u;
            asm volatile("global_load_async_to_lds_b64 %0, %1, %2"
                         :: "v"(loff), "v"(goff), "s"(Mb) : "memory");
        }
        asm volatile("s_wait_asynccnt 0" ::: "memory");
    }
    __syncthreads();

    // ---- per-row min, then K = exp(-gamma*(M - rowmin)) in place ----
    {
        float* rowp = K + tid * PITCH;     // tid == row
        float mn = rowp[0];
        for (int k = 1; k < W_; ++k) mn = fminf(mn, rowp[k]);
        for (int k = 0; k < W_; ++k) rowp[k] = __expf(-GAMMA_C * (rowp[k] - mn));
    }
    if (tid < W_) v[tid] = 1.0f;           // initial scaling vector
    __syncthreads();

    bool done = false;
    for (int it = 0; it < MAXITERS_C && !done; ++it) {
        // ---------- row sums s = K * v  (WMMA, B = v broadcast over columns) ----------
        for (int sidx = 0; sidx < 2; ++sidx) {
            const int r0 = wave * 32 + sidx * 16;
            const float* arow = K + (r0 + lo) * PITCH + 2 * hi;
            v8f acc0 = {}, acc1 = {};
            for (int k0 = 0; k0 < W_; k0 += 8) {
                v2f a0 = *(const v2f*)(arow + k0);
                v2f b0 = *(const v2f*)(v + k0 + 2 * hi);
                acc0 = __builtin_amdgcn_wmma_f32_16x16x4_f32(
                        false, a0, false, b0, (short)0, acc0, false, false);
                v2f a1 = *(const v2f*)(arow + k0 + 4);
                v2f b1 = *(const v2f*)(v + k0 + 4 + 2 * hi);
                acc1 = __builtin_amdgcn_wmma_f32_16x16x4_f32(
                        false, a1, false, b1, (short)0, acc1, false, false);
            }
            // D: VGPR q = row r0+q (lanes 0-15) / r0+8+q (lanes 16-31), repl. over N.
            if (lo == 0) {
                float* dst = u + r0 + 8 * hi;
                #pragma unroll
                for (int q = 0; q < 8; ++q) dst[q] = acc0[q] + acc1[q];
            }
        }
        __syncthreads();
        if (tid < H_) u[tid] = rmarg / u[tid];                   // u = r / (K v)
        __syncthreads();

        // ---------- col sums t = K^T * u (WMMA, A = u broadcast over rows) ----------
        for (int sidx = 0; sidx < 2; ++sidx) {
            const int c0 = wave * 32 + sidx * 16;
            v8f acc0 = {}, acc1 = {};
            for (int k0 = 0; k0 < H_; k0 += 8) {
                const int kr0 = k0 + 2 * hi;
                v2f a0 = *(const v2f*)(u + kr0);
                v2f b0;
                b0[0] = K[kr0 * PITCH + c0 + lo];
                b0[1] = K[(kr0 + 1) * PITCH + c0 + lo];
                acc0 = __builtin_amdgcn_wmma_f32_16x16x4_f32(
                        false, a0, false, b0, (short)0, acc0, false, false);
                const int kr1 = kr0 + 4;
                v2f a1 = *(const v2f*)(u + kr1);
                v2f b1;
                b1[0] = K[kr1 * PITCH + c0 + lo];
                b1[1] = K[(kr1 + 1) * PITCH + c0 + lo];
                acc1 = __builtin_amdgcn_wmma_f32_16x16x4_f32(
                        false, a1, false, b1, (short)0, acc1, false, false);
            }
            // D replicated over rows: lane lo holds column c0+lo in every VGPR.
            if (lane < 16) t[c0 + lo] = acc0[0] + acc1[0];
        }
        __syncthreads();

        // ---------- convergence test + column update ----------
        if (tid == 0) *maxbits = 0u;
        __syncthreads();
        float tj = t[tid];
        float diff = fabsf(v[tid] * tj - cmarg);                 // |beta - c|
        atomicMax(maxbits, __float_as_uint(diff));               // ds_max_u32 (diff>=0)
        __syncthreads();
        done = (__uint_as_float(*maxbits) <= EPS_C);
        if (!done) v[tid] = cmarg / tj;                          // col-normalize
        __syncthreads();
    }

    // ---- P = diag(u) K diag(v), coalesced store ----
    for (int idx = tid; idx < H_ * W_; idx += NTHREADS) {
        int row = idx >> 8, col = idx & 255;
        Pb[idx] = u[row] * K[row * PITCH + col] * v[col];
    }
}

extern "C" void kernel_launch(void* const* d_in, const int* in_sizes, int n_in,
                              void* d_out, int out_size, void* d_ws, size_t ws_size,
                              hipStream_t stream) {
    (void)n_in; (void)d_ws; (void)ws_size; (void)out_size;
    const float* M = (const float*)d_in[0];
    float* P = (float*)d_out;
    const int batch = in_sizes[0] / (H_ * W_);

    const size_t smem_bytes = (size_t)(H_ * PITCH + H_ + 2 * W_ + 4) * sizeof(float);
    (void)hipFuncSetAttribute((const void*)sinkhorn_wmma_kernel,
                              hipFuncAttributeMaxDynamicSharedMemorySize,
                              (int)smem_bytes);

    sinkhorn_wmma_kernel<<<batch, NTHREADS, smem_bytes, stream>>>(M, P);
}